// loss_func8_69621419868225
// MI455X (gfx1250) — compile-verified
//
#include <hip/hip_runtime.h>

typedef __attribute__((ext_vector_type(2))) float v2f;
typedef __attribute__((ext_vector_type(8))) float v8f;

#define NB 8
#define CB 8
#define HH 512
#define WW 512
#define TS 16

// Gaussian 5-tap, sigma=2, normalized: [g0,g1,g2,g1,g0]
#define G0f 0.15246914f
#define G1f 0.22184131f
#define G2f 0.25137913f

__device__ __forceinline__ int reflect101(int x, int n) {
    if (x < 0) x = -x;
    if (x >= n) x = 2 * n - 2 - x;
    return x;
}

// select g[d] for d in [0,5), else 0 — pure v_cndmask, no scratch
__device__ __forceinline__ float gsel(int d) {
    float r = 0.f;
    r = (d == 0 || d == 4) ? G0f : r;
    r = (d == 1 || d == 3) ? G1f : r;
    r = (d == 2) ? G2f : r;
    return r;
}

// horizontal 5-tap blur of one padded row segment (LDS)
__device__ __forceinline__ float hblur(const float* row) {
    return G0f * (row[0] + row[4]) + G1f * (row[1] + row[3]) + G2f * row[2];
}

__global__ __launch_bounds__(256, 4) void fused_loss_kernel(
    const float* __restrict__ pan, const float* __restrict__ ms,
    const float* __restrict__ outp, double* __restrict__ acc)
{
    __shared__ float P[CB][20][20];   // reflect-padded per-channel tiles of `out`
    __shared__ float mp[18][18];      // zero-padded channel-mean tile
    __shared__ float pp[18][18];      // zero-padded pan tile
    __shared__ float red[256];

    const int tid  = threadIdx.x;
    const int lane = tid & 31;
    const int wv   = tid >> 5;        // wave id == channel id
    const int bx   = blockIdx.x * TS; // col origin
    const int by   = blockIdx.y * TS; // row origin
    const int n    = blockIdx.z;

    // ---- load reflect-101 padded 20x20 tile of `out`, channel = wave ----
    {
        const float* src = outp + ((size_t)(n * CB + wv)) * (HH * WW);
        for (int i = lane; i < 400; i += 32) {
            int r = i / 20, q = i - r * 20;
            int gr = reflect101(by - 2 + r, HH);
            int gc = reflect101(bx - 2 + q, WW);
            P[wv][r][q] = src[gr * WW + gc];
        }
    }
    // ---- load zero-padded 18x18 pan tile ----
    for (int i = tid; i < 324; i += 256) {
        int r = i / 18, q = i - r * 18;
        int gr = by - 1 + r, gc = bx - 1 + q;
        pp[r][q] = (gr >= 0 && gr < HH && gc >= 0 && gc < WW)
                     ? pan[((size_t)n * HH + gr) * WW + gc] : 0.f;
    }
    __syncthreads();

    // ---- per-wave: separable 5x5 Gaussian via WMMA vertical pass ----
    // D(16x16) = sum_k A_k(16x4) x B_k(4x16);  A_k[m][j] = g[(4k+j)-m], B_k[j][c] = Hrow[4k+j][c]
    const int m    = lane & 15;       // A row; also B/D column
    const int half = lane >> 4;
    const int j0   = 2 * half;        // K index carried by VGPR0 in this lane half

    const float* msrc = ms + (((size_t)(n * CB + wv)) * HH + by) * WW + bx;
    __builtin_prefetch(msrc, 0, 1);   // global_prefetch_b8

    v8f bacc = {0.f, 0.f, 0.f, 0.f, 0.f, 0.f, 0.f, 0.f};
#pragma unroll
    for (int k = 0; k < 5; ++k) {
        const int t0 = 4 * k + j0;
        const int t1 = t0 + 1;
        v2f a, b;
        a.x = gsel(t0 - m);
        a.y = gsel(t1 - m);
        b.x = hblur(&P[wv][t0][m]);   // Hrow[t0][col=m]
        b.y = hblur(&P[wv][t1][m]);   // Hrow[t1][col=m]
        bacc = __builtin_amdgcn_wmma_f32_16x16x4_f32(
            false, a, false, b, (short)0, bacc, false, false);
    }

    // ---- L1 |blur - ms| using the f32 C/D layout (VGPR j -> row j+8*half, col m) ----
    float s_ms = 0.f;
#pragma unroll
    for (int j = 0; j < 8; ++j) {
        int row = j + 8 * half;
        s_ms += fabsf(bacc[j] - msrc[row * WW + m]);
    }

    // ---- channel-mean tile (Sobel wants zero padding outside the image) ----
    for (int i = tid; i < 324; i += 256) {
        int r = i / 18, q = i - r * 18;
        int gr = by - 1 + r, gc = bx - 1 + q;
        float s = 0.f;
        if (gr >= 0 && gr < HH && gc >= 0 && gc < WW) {
#pragma unroll
            for (int c2 = 0; c2 < CB; ++c2) s += P[c2][r + 1][q + 1];
            s *= 0.125f;
        }
        mp[r][q] = s;
    }
    __syncthreads();

    // ---- Sobel of mean-tile vs Sobel of pan, L1 ----
    float s_pan;
    {
        const int r = tid >> 4, q = tid & 15;
        float o00 = mp[r][q],     o01 = mp[r][q + 1],     o02 = mp[r][q + 2];
        float o10 = mp[r + 1][q],                          o12 = mp[r + 1][q + 2];
        float o20 = mp[r + 2][q], o21 = mp[r + 2][q + 1], o22 = mp[r + 2][q + 2];
        float ogx = (o02 - o00) + 2.f * (o12 - o10) + (o22 - o20);
        float ogy = (o20 - o00) + 2.f * (o21 - o01) + (o22 - o02);

        float p00 = pp[r][q],     p01 = pp[r][q + 1],     p02 = pp[r][q + 2];
        float p10 = pp[r + 1][q],                          p12 = pp[r + 1][q + 2];
        float p20 = pp[r + 2][q], p21 = pp[r + 2][q + 1], p22 = pp[r + 2][q + 2];
        float pgx = (p02 - p00) + 2.f * (p12 - p10) + (p22 - p20);
        float pgy = (p20 - p00) + 2.f * (p21 - p01) + (p22 - p02);

        s_pan = fabsf(ogx - pgx) + fabsf(ogy - pgy);
    }

    // ---- block reduction -> double atomics ----
    red[tid] = s_ms;
    __syncthreads();
    for (int s = 128; s > 0; s >>= 1) {
        if (tid < s) red[tid] += red[tid + s];
        __syncthreads();
    }
    if (tid == 0) atomicAdd(&acc[0], (double)red[0]);
    __syncthreads();

    red[tid] = s_pan;
    __syncthreads();
    for (int s = 128; s > 0; s >>= 1) {
        if (tid < s) red[tid] += red[tid + s];
        __syncthreads();
    }
    if (tid == 0) atomicAdd(&acc[1], (double)red[0]);
}

__global__ void init_acc_kernel(double* acc) {
    acc[0] = 0.0;
    acc[1] = 0.0;
}

__global__ void finalize_kernel(const double* __restrict__ acc, float* __restrict__ out) {
    // mean over (8,8,512,512) and (8,1,512,512)
    out[0] = (float)(acc[0] * (1.0 / 16777216.0) + acc[1] * (1.0 / 2097152.0));
}

extern "C" void kernel_launch(void* const* d_in, const int* in_sizes, int n_in,
                              void* d_out, int out_size, void* d_ws, size_t ws_size,
                              hipStream_t stream) {
    const float* pan = (const float*)d_in[0];  // (8,1,512,512)
    const float* ms  = (const float*)d_in[1];  // (8,8,512,512)
    const float* out = (const float*)d_in[2];  // (8,8,512,512)
    float* loss = (float*)d_out;
    double* acc = (double*)d_ws;

    init_acc_kernel<<<1, 1, 0, stream>>>(acc);
    dim3 grid(WW / TS, HH / TS, NB);   // 32 x 32 x 8
    fused_loss_kernel<<<grid, 256, 0, stream>>>(pan, ms, out, acc);
    finalize_kernel<<<1, 1, 0, stream>>>(acc, loss);
}